// Logm_54872502174210
// MI455X (gfx1250) — compile-verified
//
#include <hip/hip_runtime.h>
#include <math.h>

#define N_PER_BATCH (64 * 128 * 128)   // 2^20 voxels per batch
#define THREADS 256
#define VPT 4                           // voxels per thread (float4)

typedef __attribute__((ext_vector_type(4))) float v4f;

__device__ __forceinline__ float f4get(const float4& f, int j) {
    switch (j) { case 0: return f.x; case 1: return f.y; case 2: return f.z; default: return f.w; }
}
__device__ __forceinline__ void f4set(float4& f, int j, float v) {
    switch (j) { case 0: f.x = v; break; case 1: f.y = v; break; case 2: f.z = v; break; default: f.w = v; }
}

// Branchless acos for x in [-1,1]; Abramowitz-Stegun 4.4.46 (|err| <= 2e-8 rad)
__device__ __forceinline__ float fast_acos(float x) {
    const float ax = fabsf(x);
    float p = -0.0012624911f;
    p = fmaf(p, ax,  0.0066700901f);
    p = fmaf(p, ax, -0.0170881256f);
    p = fmaf(p, ax,  0.0308918810f);
    p = fmaf(p, ax, -0.0501743046f);
    p = fmaf(p, ax,  0.0889789874f);
    p = fmaf(p, ax, -0.2145988016f);
    p = fmaf(p, ax,  1.5707963050f);
    const float r = __builtin_amdgcn_sqrtf(1.0f - ax) * p;
    return (x >= 0.0f) ? r : (3.14159265358979f - r);
}

__global__ __launch_bounds__(THREADS) void logm3x3_kernel(const float* __restrict__ x,
                                                          float* __restrict__ y)
{
    // LDS staging: 9 channels x 256 threads x float4 = 36 KB
    __shared__ float4 tile[9 * THREADS];

    const unsigned t  = threadIdx.x;
    const unsigned g4 = (blockIdx.x * THREADS + t) * VPT;   // first voxel of this thread's group
    const unsigned b  = g4 >> 20;                           // batch index (n = 2^20)
    const unsigned v  = g4 & (N_PER_BATCH - 1);             // voxel within batch

    const unsigned ldsBase = (unsigned)(size_t)tile;        // LDS byte offset of tile[0]

    // --- CDNA5 async staging: 9 x global_load_async_to_lds_b128, one wait ---
    #pragma unroll
    for (int c = 0; c < 9; ++c) {
        unsigned gByte = (((b * 9u + (unsigned)c) << 20) + v) * 4u;        // byte offset into x
        unsigned lByte = ldsBase + ((unsigned)c * THREADS + t) * 16u;      // LDS dest byte offset
        asm volatile("global_load_async_to_lds_b128 %0, %1, %2"
                     :: "v"(lByte), "v"(gByte), "s"(x) : "memory");
    }
    asm volatile("s_wait_asynccnt 0" ::: "memory");

    float4 m[9];
    #pragma unroll
    for (int c = 0; c < 9; ++c) m[c] = tile[c * THREADS + t];

    float4 r[9];

    #pragma unroll
    for (int j = 0; j < VPT; ++j) {
        const float a00 = f4get(m[0], j);
        const float a01 = f4get(m[1], j);
        const float a02 = f4get(m[2], j);
        const float a11 = f4get(m[4], j);
        const float a12 = f4get(m[5], j);
        const float a22 = f4get(m[8], j);

        // ---- closed-form eigenvalues of symmetric 3x3 (spectrum >= 3 by construction) ----
        const float q   = (a00 + a11 + a22) * (1.0f / 3.0f);
        const float b00 = a00 - q, b11 = a11 - q, b22 = a22 - q;
        const float off2 = a01 * a01 + a02 * a02 + a12 * a12;
        const float p2  = b00 * b00 + b11 * b11 + b22 * b22 + 2.0f * off2;
        const float pp  = fmaxf(p2 * (1.0f / 6.0f), 1.0e-30f);
        const float p   = __builtin_amdgcn_sqrtf(pp);
        const float ip  = __builtin_amdgcn_rsqf(pp);
        const float detB = b00 * (b11 * b22 - a12 * a12)
                         - a01 * (a01 * b22 - a12 * a02)
                         + a02 * (a01 * a12 - b11 * a02);
        float rr = 0.5f * detB * ip * ip * ip;
        rr = fminf(1.0f, fmaxf(-1.0f, rr));
        const float phi = fast_acos(rr) * (1.0f / 3.0f);
        const float l1  = q + 2.0f * p * __cosf(phi);                       // largest
        const float l3  = q + 2.0f * p * __cosf(phi + 2.0943951023931953f); // smallest
        const float l2  = 3.0f * q - l1 - l3;

        const float ll1 = __logf(l1);
        const float ll2 = __logf(l2);
        const float ll3 = __logf(l3);

        // ---- general spectral-projector path (polynomials in M; branchless guards) ----
        const float m200 = a00 * a00 + a01 * a01 + a02 * a02;
        const float m201 = a00 * a01 + a01 * a11 + a02 * a12;
        const float m202 = a00 * a02 + a01 * a12 + a02 * a22;
        const float m211 = a01 * a01 + a11 * a11 + a12 * a12;
        const float m212 = a01 * a02 + a11 * a12 + a12 * a22;
        const float m222 = a02 * a02 + a12 * a12 + a22 * a22;

        // guarded log divided differences (ternary -> v_cndmask, no control flow)
        const float d12 = l1 - l2, d32 = l3 - l2;
        const float g12 = (fabsf(d12) > 1.0e-6f * (l1 + l2))
                            ? (ll1 - ll2) * __builtin_amdgcn_rcpf(d12)
                            : 2.0f * __builtin_amdgcn_rcpf(l1 + l2);
        const float g32 = (fabsf(d32) > 1.0e-6f * (l3 + l2))
                            ? (ll3 - ll2) * __builtin_amdgcn_rcpf(d32)
                            : 2.0f * __builtin_amdgcn_rcpf(l3 + l2);
        const float inv13 = __builtin_amdgcn_rcpf(fmaxf(l1 - l3, 1.0e-12f));
        const float c1 =  g12 * inv13;   // coeff of (M^2 - (l2+l3)M + l2 l3 I)
        const float c3 = -g32 * inv13;   // coeff of (M^2 - (l1+l2)M + l1 l2 I)
        const float s1 = l2 + l3, e1 = l2 * l3;
        const float s3 = l1 + l2, e3 = l1 * l2;

        float y00 = ll2 + c1 * (m200 - s1 * a00 + e1) + c3 * (m200 - s3 * a00 + e3);
        float y01 =       c1 * (m201 - s1 * a01)      + c3 * (m201 - s3 * a01);
        float y02 =       c1 * (m202 - s1 * a02)      + c3 * (m202 - s3 * a02);
        float y11 = ll2 + c1 * (m211 - s1 * a11 + e1) + c3 * (m211 - s3 * a11 + e3);
        float y12 =       c1 * (m212 - s1 * a12)      + c3 * (m212 - s3 * a12);
        float y22 = ll2 + c1 * (m222 - s1 * a22 + e1) + c3 * (m222 - s3 * a22 + e3);

        // ---- near-spherical fallback (first order about l2*I), branchless select ----
        const bool deg = (l1 - l3) <= 1.0e-6f * l2;
        const float inv = __builtin_amdgcn_rcpf(l2);
        y00 = deg ? (ll2 + (a00 - l2) * inv) : y00;
        y11 = deg ? (ll2 + (a11 - l2) * inv) : y11;
        y22 = deg ? (ll2 + (a22 - l2) * inv) : y22;
        y01 = deg ? (a01 * inv) : y01;
        y02 = deg ? (a02 * inv) : y02;
        y12 = deg ? (a12 * inv) : y12;

        f4set(r[0], j, y00); f4set(r[1], j, y01); f4set(r[2], j, y02);
        f4set(r[3], j, y01); f4set(r[4], j, y11); f4set(r[5], j, y12);
        f4set(r[6], j, y02); f4set(r[7], j, y12); f4set(r[8], j, y22);
    }

    // streamed output: non-temporal B128 stores (native vector type for the builtin)
    #pragma unroll
    for (int c = 0; c < 9; ++c) {
        v4f* dst = (v4f*)(y + ((size_t)(b * 9u + (unsigned)c) << 20) + v);
        v4f val;
        val.x = r[c].x; val.y = r[c].y; val.z = r[c].z; val.w = r[c].w;
        __builtin_nontemporal_store(val, dst);
    }
}

extern "C" void kernel_launch(void* const* d_in, const int* in_sizes, int n_in,
                              void* d_out, int out_size, void* d_ws, size_t ws_size,
                              hipStream_t stream) {
    (void)n_in; (void)out_size; (void)d_ws; (void)ws_size;
    const float* x = (const float*)d_in[0];
    float*       y = (float*)d_out;

    const int total_vox = in_sizes[0] / 9;              // B * D * H * W = 2,097,152
    const int groups    = total_vox / VPT;              // threads needed
    const int blocks    = groups / THREADS;             // 2048
    logm3x3_kernel<<<blocks, THREADS, 0, stream>>>(x, y);
}